// KNNEmbeddingV2_55164559949911
// MI455X (gfx1250) — compile-verified
//
#include <hip/hip_runtime.h>
#include <math.h>

#define B_ 8
#define N_ 2048
#define D_ 16

typedef __attribute__((ext_vector_type(2))) float v2f;
typedef __attribute__((ext_vector_type(8))) float v8f;

static __device__ __forceinline__ v8f wmma4(v2f a, v2f b, v8f c) {
  // V_WMMA_F32_16X16X4_F32 : D = A(16x4) * B(4x16) + C, fp32 matrix pipe
  return __builtin_amdgcn_wmma_f32_16x16x4_f32(false, a, false, b, (short)0, c,
                                               false, false);
}

// Async global->LDS copy (CDNA5, ASYNCcnt-tracked). Inline asm per ISA 08:
// dst = LDS byte offset (generic-pointer low 32 bits), src = 64-bit vaddr.
static __device__ __forceinline__ void async_copy_b128(void* lds,
                                                       const void* glb) {
  unsigned l = (unsigned)(size_t)lds;
  asm volatile("global_load_async_to_lds_b128 %0, %1, off" ::"v"(l), "v"(glb)
               : "memory");
}
static __device__ __forceinline__ void async_copy_b32(void* lds,
                                                      const void* glb) {
  unsigned l = (unsigned)(size_t)lds;
  asm volatile("global_load_async_to_lds_b32 %0, %1, off" ::"v"(l), "v"(glb)
               : "memory");
}
static __device__ __forceinline__ void wait_async0() {
  asm volatile("s_wait_asynccnt 0x0" ::: "memory");
}

// ---------------------------------------------------------------------------
// Kernel A: per-batch per-channel (32 channels) mean & 1/(std+1e-5), ddof=1
// ---------------------------------------------------------------------------
__global__ __launch_bounds__(256) void k_stats(const float* __restrict__ x,
                                               const float* __restrict__ feat,
                                               float* __restrict__ meanv,
                                               float* __restrict__ invv) {
  int b = blockIdx.x;
  int t = threadIdx.x;
  int c = t & 31;    // channel 0..31
  int grp = t >> 5;  // 0..7
  int d = c & 15;
  bool is_ftr = (c >= 16);
  bool mask = feat[b * D_ + d] > 0.1f;  // true -> value goes to ftr half
  float s1 = 0.f, s2 = 0.f;
  for (int r = 0; r < N_ / 8; ++r) {
    int n = grp + (r << 3);
    float v = x[((size_t)b * N_ + n) * D_ + d];
    float xc = (mask == is_ftr) ? v : 0.f;
    s1 += xc;
    s2 += xc * xc;
  }
  __shared__ float l1[256];
  __shared__ float l2[256];
  l1[t] = s1;
  l2[t] = s2;
  __syncthreads();
  if (t < 32) {
    float a1 = 0.f, a2 = 0.f;
#pragma unroll
    for (int g = 0; g < 8; ++g) {
      a1 += l1[t + 32 * g];
      a2 += l2[t + 32 * g];
    }
    float mean = a1 / (float)N_;
    float var = (a2 - (float)N_ * mean * mean) / (float)(N_ - 1);
    var = fmaxf(var, 0.f);
    meanv[b * 32 + t] = mean;
    invv[b * 32 + t] = 1.f / (sqrtf(var) + 1e-5f);
  }
}

// ---------------------------------------------------------------------------
// Kernel B: x_crd rows, sq = |x_crd|^2, R_crd/R_ftr = sums of clipped norm
// ---------------------------------------------------------------------------
__global__ __launch_bounds__(256) void k_rows(
    const float* __restrict__ x, const float* __restrict__ feat,
    const float* __restrict__ meanv, const float* __restrict__ invv,
    float* __restrict__ Xc, float* __restrict__ sqv, float* __restrict__ Rc,
    float* __restrict__ Rf) {
  __shared__ float sf[16];
  __shared__ float sm[32];
  __shared__ float si[32];
  int b = blockIdx.x / (N_ / 256);
  int n0 = (blockIdx.x % (N_ / 256)) * 256;
  int t = threadIdx.x;
  if (t < 16)
    sf[t] = feat[b * 16 + t];
  else if (t < 48)
    sm[t - 16] = meanv[b * 32 + (t - 16)];
  else if (t < 80)
    si[t - 48] = invv[b * 32 + (t - 48)];
  __syncthreads();
  size_t row = (size_t)b * N_ + (n0 + t);
  float xv[16];
#pragma unroll
  for (int d4 = 0; d4 < 4; ++d4) {
    float4 v = *(const float4*)&x[row * 16 + d4 * 4];
    xv[d4 * 4 + 0] = v.x;
    xv[d4 * 4 + 1] = v.y;
    xv[d4 * 4 + 2] = v.z;
    xv[d4 * 4 + 3] = v.w;
  }
  float sq = 0.f, rc = 0.f, rf = 0.f;
#pragma unroll
  for (int d = 0; d < 16; ++d) {
    bool mask = sf[d] > 0.1f;
    float xc = mask ? 0.f : xv[d];
    float xf = mask ? xv[d] : 0.f;
    sq += xc * xc;
    float nc = fminf(fmaxf((xc - sm[d]) * si[d], -10.f), 10.f);
    float nf = fminf(fmaxf((xf - sm[16 + d]) * si[16 + d], -10.f), 10.f);
    rc += nc;
    rf += nf;
    xv[d] = xc;
  }
#pragma unroll
  for (int d4 = 0; d4 < 4; ++d4) {
    float4 v = make_float4(xv[d4 * 4 + 0], xv[d4 * 4 + 1], xv[d4 * 4 + 2],
                           xv[d4 * 4 + 3]);
    *(float4*)&Xc[row * 16 + d4 * 4] = v;
  }
  sqv[row] = sq;
  Rc[row] = rc;
  Rf[row] = rf;
}

// ---------------------------------------------------------------------------
// Kernel C: brute-force KNN (top-16 smallest d2, index tiebreak), emit S.
// Candidate chunks double-buffered in LDS via async global->LDS copies so the
// next chunk streams in behind the FMA + insertion work (ASYNCcnt-tracked).
// ---------------------------------------------------------------------------
__global__ __launch_bounds__(256) void k_knn(
    const float* __restrict__ Xc, const float* __restrict__ sqv,
    const float* __restrict__ Rc, const float* __restrict__ Rf,
    float* __restrict__ Sc, float* __restrict__ Sf) {
  __shared__ float Xj[2][256 * 16];
  __shared__ float sqj[2][256];
  int t = threadIdx.x;
  int b = blockIdx.x >> 3;
  int i = ((blockIdx.x & 7) << 8) + t;
  size_t rb = (size_t)b * N_;
  float xi[16];
#pragma unroll
  for (int d4 = 0; d4 < 4; ++d4) {
    float4 v = *(const float4*)&Xc[(rb + i) * 16 + d4 * 4];
    xi[d4 * 4 + 0] = v.x;
    xi[d4 * 4 + 1] = v.y;
    xi[d4 * 4 + 2] = v.z;
    xi[d4 * 4 + 3] = v.w;
  }
  float sqi = sqv[rb + i];
  float dist[16];
  int idxl[16];
#pragma unroll
  for (int k = 0; k < 16; ++k) {
    dist[k] = 3.4e38f;
    idxl[k] = 0;
  }
  // Prefetch chunk 0 into buffer 0.
  async_copy_b128(&Xj[0][t * 16], &Xc[(rb + t) * 16]);
  async_copy_b32(&sqj[0][t], &sqv[rb + t]);

  int buf = 0;
  for (int c0 = 0; c0 < N_; c0 += 256, buf ^= 1) {
    wait_async0();
    __syncthreads();  // chunk c0 resident in Xj[buf] for all waves
    if (c0 + 256 < N_) {
      // Prefetch next chunk into the other buffer (its last readers finished
      // before the barrier above).
      async_copy_b128(&Xj[buf ^ 1][t * 16], &Xc[(rb + c0 + 256 + t) * 16]);
      async_copy_b32(&sqj[buf ^ 1][t], &sqv[rb + c0 + 256 + t]);
    }
    const float* Xrow = &Xj[buf][0];
    const float* Sq = &sqj[buf][0];
    for (int jj = 0; jj < 256; ++jj) {
      float dot = 0.f;
#pragma unroll
      for (int d = 0; d < 16; ++d) dot = fmaf(xi[d], Xrow[jj * 16 + d], dot);
      float d2 = fmaxf(sqi + Sq[jj] - 2.f * dot, 0.f);
      if (d2 < dist[15]) {
        float dd = d2;
        int jn = c0 + jj;
#pragma unroll
        for (int s = 0; s < 16; ++s) {
          if (dd < dist[s]) {
            float td = dist[s];
            int ti = idxl[s];
            dist[s] = dd;
            idxl[s] = jn;
            dd = td;
            jn = ti;
          }
        }
      }
    }
  }
  float ric = Rc[rb + i], rif = Rf[rb + i];
#pragma unroll
  for (int k = 0; k < 16; ++k) {
    int jn = idxl[k];
    Sc[(rb + i) * 16 + k] = Rc[rb + jn] - ric;
    Sf[(rb + i) * 16 + k] = Rf[rb + jn] - rif;
  }
}

// ---------------------------------------------------------------------------
// Kernel D: fused  h = S @ Wcat^T (16x32 @ 32x1024) -> GLU -> out = g @ Wout^T
// One workgroup = 16 rows; 8 waves; fp32 WMMA 16x16x4 throughout.
// ---------------------------------------------------------------------------
__global__ __launch_bounds__(256) void k_gemm(
    const float* __restrict__ Sc, const float* __restrict__ Sf,
    const float* __restrict__ Wcrd, const float* __restrict__ Wftr,
    const float* __restrict__ Wout, float* __restrict__ out) {
  __shared__ float Ssh[16 * 34];   // S tile, padded (even stride, no conflicts)
  __shared__ float Gsh[16 * 516];  // g tile (16 x 512), padded
  int b = blockIdx.x >> 7;
  int i0 = (blockIdx.x & 127) << 4;
  size_t rb = (size_t)b * N_ + i0;
  int t = threadIdx.x;
  {
    int r = t >> 4;  // 0..15
    int k = t & 15;  // 0..15
    Ssh[r * 34 + k] = Sc[(rb + r) * 16 + k];
    Ssh[r * 34 + 16 + k] = Sf[(rb + r) * 16 + k];
  }
  __syncthreads();
  int lane = t & 31, wave = t >> 5;
  int half = lane >> 4, lm = lane & 15;

  // ---- GEMM1 + GLU: each wave owns 4 (a,b) column-tile pairs of h ----
  for (int p = 0; p < 4; ++p) {
    int f0 = (wave * 4 + p) * 16;  // a-cols f0..f0+15 ; b-cols f0+512..
    v8f accA = {0, 0, 0, 0, 0, 0, 0, 0};
    v8f accB = {0, 0, 0, 0, 0, 0, 0, 0};
#pragma unroll
    for (int k0 = 0; k0 < 32; k0 += 4) {
      int kk = k0 + 2 * half;
      // A fragment: lane holds S[row=lm, kk..kk+1]
      v2f af = *(const v2f*)&Ssh[lm * 34 + kk];
      // B fragment: lane holds Wcat^T[kk..kk+1, f] = Wsel[f, kk..kk+1]
      const float* Wp = (k0 < 16) ? Wcrd : Wftr;
      int kw = kk & 15;
      int fa = f0 + lm;
      v2f bA = *(const v2f*)&Wp[fa * 16 + kw];
      v2f bB = *(const v2f*)&Wp[(fa + 512) * 16 + kw];
      accA = wmma4(af, bA, accA);
      accB = wmma4(af, bB, accB);
    }
#pragma unroll
    for (int r = 0; r < 8; ++r) {
      float a = accA[r];
      float bb = accB[r];
      // sigmoid via v_exp + v_rcp (no IEEE-division sequence needed)
      float g = a * __builtin_amdgcn_rcpf(1.f + __expf(-bb));
      int row = r + 8 * half;  // C/D layout: vgpr r, lane-half selects +8
      Gsh[row * 516 + f0 + lm] = g;
    }
  }
  __syncthreads();

  // ---- GEMM2: out(16x256) = g(16x512) @ Wout^T(512x256); 2 m-tiles/wave ----
  for (int q = 0; q < 2; ++q) {
    int m0 = (wave * 2 + q) * 16;
    v8f acc = {0, 0, 0, 0, 0, 0, 0, 0};
#pragma unroll 8
    for (int k0 = 0; k0 < 512; k0 += 4) {
      int f = k0 + 2 * half;
      v2f af = *(const v2f*)&Gsh[lm * 516 + f];
      v2f bf = *(const v2f*)&Wout[(size_t)(m0 + lm) * 512 + f];
      acc = wmma4(af, bf, acc);
    }
#pragma unroll
    for (int r = 0; r < 8; ++r) {
      int row = i0 + r + 8 * half;
      out[(((size_t)b * N_ + row) << 8) + m0 + lm] = acc[r];
    }
  }
}

extern "C" void kernel_launch(void* const* d_in, const int* in_sizes, int n_in,
                              void* d_out, int out_size, void* d_ws,
                              size_t ws_size, hipStream_t stream) {
  const float* x = (const float*)d_in[0];
  const float* feat = (const float*)d_in[1];
  const float* Wcrd = (const float*)d_in[2];
  const float* Wftr = (const float*)d_in[3];
  const float* Wout = (const float*)d_in[4];
  float* out = (float*)d_out;

  float* ws = (float*)d_ws;
  float* meanv = ws;          // 256
  float* invv = meanv + 256;  // 256
  float* Xc = invv + 256;     // 8*2048*16 = 262144
  float* sqv = Xc + 262144;   // 16384
  float* Rc = sqv + 16384;    // 16384
  float* Rf = Rc + 16384;     // 16384
  float* Sc = Rf + 16384;     // 262144
  float* Sf = Sc + 262144;    // 262144

  k_stats<<<8, 256, 0, stream>>>(x, feat, meanv, invv);
  k_rows<<<64, 256, 0, stream>>>(x, feat, meanv, invv, Xc, sqv, Rc, Rf);
  k_knn<<<64, 256, 0, stream>>>(Xc, sqv, Rc, Rf, Sc, Sf);
  k_gemm<<<1024, 256, 0, stream>>>(Sc, Sf, Wcrd, Wftr, Wout, out);
}